// MultiHeadAttentionLayer_39161511805520
// MI455X (gfx1250) — compile-verified
//
#include <hip/hip_runtime.h>

// Problem constants
#define BB   4
#define LL   256
#define HIDD 512
#define NHH  8
#define DD   64
#define KSTRIDE 66   // 64 data dwords + 2 pad dwords -> bank (2k+d)%64, conflict-free

typedef __attribute__((ext_vector_type(2))) float v2f;
typedef __attribute__((ext_vector_type(8))) float v8f;
typedef __attribute__((ext_vector_type(4))) unsigned int v4u;
typedef __attribute__((ext_vector_type(8))) int v8i;
typedef __attribute__((ext_vector_type(4))) int v4i;
typedef int v2i_g __attribute__((vector_size(2 * sizeof(int))));  // matches builtin param

#if __has_builtin(__builtin_amdgcn_tensor_load_to_lds) && \
    __has_builtin(__builtin_amdgcn_s_wait_tensorcnt)
#define HAVE_TDM 1
#endif
#if __has_include(<hip/amd_detail/amd_gfx1250_TDM.h>)
#define TDM_SIX_ARGS 1
#endif
#if __has_builtin(__builtin_amdgcn_global_load_async_to_lds_b64)
#define HAVE_ASYNC 1
#endif

// tanh(x) = 1 - 2/(exp(2x)+1) via exp2/rcp TRANS ops; saturates correctly at +/-inf.
__device__ __forceinline__ float fast_tanh(float x) {
  float t = __builtin_amdgcn_exp2f(x * 2.8853900817779268f); // 2*log2(e)
  return 1.0f - 2.0f * __builtin_amdgcn_rcpf(t + 1.0f);
}

// ---------------------------------------------------------------------------
// Generic C[m][n] = sum_k A[m][k] * W[n][k] + bias[n] using V_WMMA_F32_16X16X4_F32.
// One wave computes one 16x16 tile of C; K-loop steps by 4.
// out_mode 0: scatter to head-split [B,H,L,D];  out_mode 1: plain row-major [M][N].
// ---------------------------------------------------------------------------
__global__ void wmma_gemm_nt(const float* __restrict__ A, int lda,
                             const float* __restrict__ W,
                             const float* __restrict__ bias,
                             float* __restrict__ Cout,
                             int M, int N, int K, int out_mode)
{
  const int lane = threadIdx.x & 31;
  const int wave = threadIdx.x >> 5;
  const int tile = blockIdx.x * (blockDim.x >> 5) + wave;
  const int ntn  = N >> 4;
  const int tm   = tile / ntn;
  const int tn   = tile % ntn;
  const int half = lane >> 4;   // selects k-pair
  const int l16  = lane & 15;   // m for A-frag, n for B-frag

  const float* arow = A + (size_t)(tm * 16 + l16) * lda + 2 * half;
  const float* wrow = W + (size_t)(tn * 16 + l16) * K   + 2 * half;

  __builtin_prefetch(arow, 0, 0);   // global_prefetch_b8
  __builtin_prefetch(wrow, 0, 0);

  v8f acc = {};
  for (int kk = 0; kk < K; kk += 4) {
    v2f a = *(const v2f*)(arow + kk);
    v2f b = *(const v2f*)(wrow + kk);
    acc = __builtin_amdgcn_wmma_f32_16x16x4_f32(
        /*neg_a=*/false, a, /*neg_b=*/false, b,
        /*c_mod=*/(short)0, acc, /*reuse_a=*/false, /*reuse_b=*/false);
  }

  const int   col = tn * 16 + l16;
  const float bv  = bias[col];
#pragma unroll
  for (int r = 0; r < 8; ++r) {
    const int row = tm * 16 + r + half * 8;   // C/D layout: rows r and r+8
    const float val = acc[r] + bv;
    size_t idx;
    if (out_mode == 0) {
      idx = (((size_t)(row >> 8) * NHH + (col >> 6)) * LL + (row & 255)) * DD + (col & 63);
    } else {
      idx = (size_t)row * N + col;
    }
    Cout[idx] = val;
  }
}

// ---------------------------------------------------------------------------
// energy[q,k] = sum_d Vw[d]*tanh(Qe[q,d]+Ke[k,d]) + Vb, then softmax over k.
// Block = (b,h) + 32-row q chunk.  Ke tile DMA'd to LDS via TDM with pad
// (64 dwords data + 2 dwords pad == stride 66); Qe via async global->LDS.
// ---------------------------------------------------------------------------
__global__ void energy_softmax(const float* __restrict__ Qe,
                               const float* __restrict__ Ke,
                               const float* __restrict__ Vw,
                               const float* __restrict__ Vb,
                               float* __restrict__ attn)
{
  __shared__ float ke_s[LL * KSTRIDE];   // 256 x (64+2)
  __shared__ float qe_s[32 * DD];
  __shared__ float v_s[DD];

  const int tid = threadIdx.x;
  const int bh  = blockIdx.x >> 3;         // 0..31
  const int q0  = (blockIdx.x & 7) * 32;   // q chunk base
  const float* Keb = Ke + (size_t)bh * LL * DD;
  const float* Qeb = Qe + (size_t)bh * LL * DD + (size_t)q0 * DD;

  // ---- Ke tile: TDM tensor DMA (pads 2 dwords after every 64 dwords) ----
#if defined(HAVE_TDM)
  if (tid < 32) {
    unsigned long long ga = (unsigned long long)(uintptr_t)Keb;
    unsigned int       la = (unsigned int)(uintptr_t)ke_s;   // low 32 bits = LDS offset
    // D# group0 (ISA 8.3): count=1 | lds_addr | global_addr[56:0] | type=2
    v4u g0 = { 1u,
               la,
               (unsigned int)ga,
               (unsigned int)((ga >> 32) & 0x1FFFFFFu) | (2u << 30) };
    // D# group1 (ISA 8.4): data_size=2(4B), pad_enable, pad_interval=5(64dw),
    // pad_amount=1(2dw); tensor_dim0=64, tensor_dim1=256; tile 64x256; stride0=64.
    v8i g1 = { (int)((2u << 16) | (1u << 20) | (5u << 22) | (1u << 25)),
               (int)(64u << 16),            // tensor_dim0[15:0] in bits 63:48
               (int)(256u << 16),           // tensor_dim0 hi=0 | tensor_dim1[15:0]
               (int)(64u << 16),            // tensor_dim1 hi=0 | tile_dim0=64
               (int)(256u),                 // tile_dim1=256, tile_dim2=0
               (int)(64u),                  // tensor_dim0_stride = 64
               0, 0 };
    v4i gz = { 0, 0, 0, 0 };
#if defined(TDM_SIX_ARGS)
    v8i gz8 = { 0, 0, 0, 0, 0, 0, 0, 0 };
    __builtin_amdgcn_tensor_load_to_lds(g0, g1, gz, gz, gz8, 0);
#else
    __builtin_amdgcn_tensor_load_to_lds(g0, g1, gz, gz, 0);
#endif
    __builtin_amdgcn_s_wait_tensorcnt(0);
  }
#else
  for (int i = tid; i < LL * DD; i += 256)
    ke_s[(i >> 6) * KSTRIDE + (i & 63)] = Keb[i];
#endif

  // ---- Qe chunk: async global->LDS (ASYNCcnt), coalesced b64 ----
#if defined(HAVE_ASYNC)
  for (int i = tid; i < (32 * DD) / 2; i += 256) {
    __builtin_amdgcn_global_load_async_to_lds_b64(
        (v2i_g*)(Qeb + 2 * i),
        (v2i_g*)(qe_s + 2 * i), 0, 0);
  }
#if __has_builtin(__builtin_amdgcn_s_wait_asynccnt)
  __builtin_amdgcn_s_wait_asynccnt(0);
#else
  asm volatile("s_wait_asynccnt 0x0" ::: "memory");
#endif
#else
  for (int i = tid; i < 32 * DD; i += 256) qe_s[i] = Qeb[i];
#endif

  if (tid < DD) v_s[tid] = Vw[tid];
  __syncthreads();

  const float vb   = Vb[0];
  const int   lane = tid & 31;
  const int   wave = tid >> 5;

  for (int it = 0; it < 4; ++it) {
    const int ql = wave + 8 * it;          // local q row
    const float* qe = &qe_s[ql * DD];

    float acc[8];
#pragma unroll
    for (int j = 0; j < 8; ++j) acc[j] = 0.0f;

#pragma unroll 4
    for (int d2 = 0; d2 < DD / 2; ++d2) {  // ds_load_b64 pairs; conflict-free
      const v2f vv = *(const v2f*)&v_s[2 * d2];
      const v2f vq = *(const v2f*)&qe[2 * d2];
#pragma unroll
      for (int j = 0; j < 8; ++j) {
        const int k = lane + 32 * j;
        const v2f kv = *(const v2f*)&ke_s[k * KSTRIDE + 2 * d2];
        acc[j] = fmaf(vv.x, fast_tanh(vq.x + kv.x), acc[j]);
        acc[j] = fmaf(vv.y, fast_tanh(vq.y + kv.y), acc[j]);
      }
    }

    float e[8];
#pragma unroll
    for (int j = 0; j < 8; ++j) e[j] = acc[j] + vb;

    // wave-wide softmax over 256 energies
    float m = e[0];
#pragma unroll
    for (int j = 1; j < 8; ++j) m = fmaxf(m, e[j]);
#pragma unroll
    for (int off = 16; off >= 1; off >>= 1)
      m = fmaxf(m, __shfl_xor(m, off, 32));

    float s = 0.0f, p[8];
#pragma unroll
    for (int j = 0; j < 8; ++j) {
      p[j] = __builtin_amdgcn_exp2f((e[j] - m) * 1.4426950408889634f);
      s += p[j];
    }
#pragma unroll
    for (int off = 16; off >= 1; off >>= 1)
      s += __shfl_xor(s, off, 32);

    const float inv = __builtin_amdgcn_rcpf(s);
    float* arow = attn + ((size_t)bh * LL + q0 + ql) * LL;
#pragma unroll
    for (int j = 0; j < 8; ++j)
      arow[lane + 32 * j] = p[j] * inv;
  }
}

// ---------------------------------------------------------------------------
// X[b,l,h*D+d] = sum_k attn[b,h,l,k] * V[b,h,k,d]   (per-head 256x64x256 GEMM)
// ---------------------------------------------------------------------------
__global__ void attn_v(const float* __restrict__ attn,
                       const float* __restrict__ V,
                       float* __restrict__ X)
{
  const int lane = threadIdx.x & 31;
  const int wave = threadIdx.x >> 5;
  const int tile = blockIdx.x * (blockDim.x >> 5) + wave;
  const int bh = tile >> 6;           // 16 q-tiles * 4 d-tiles per (b,h)
  const int tq = (tile & 63) >> 2;
  const int td = tile & 3;
  const int half = lane >> 4;
  const int l16  = lane & 15;

  const float* arow = attn + ((size_t)bh * LL + tq * 16 + l16) * LL + 2 * half;
  const float* vhd  = V + (size_t)bh * LL * DD;
  const int dcol = td * 16 + l16;

  v8f acc = {};
  for (int kk = 0; kk < LL; kk += 4) {
    v2f a = *(const v2f*)(arow + kk);
    v2f b;
    b.x = vhd[(size_t)(kk + 2 * half)     * DD + dcol];
    b.y = vhd[(size_t)(kk + 2 * half + 1) * DD + dcol];
    acc = __builtin_amdgcn_wmma_f32_16x16x4_f32(false, a, false, b,
                                                (short)0, acc, false, false);
  }

  const int b_ = bh >> 3, h = bh & 7;
#pragma unroll
  for (int r = 0; r < 8; ++r) {
    const int q = tq * 16 + r + half * 8;
    X[((size_t)b_ * LL + q) * HIDD + h * DD + dcol] = acc[r];
  }
}

// ---------------------------------------------------------------------------
extern "C" void kernel_launch(void* const* d_in, const int* in_sizes, int n_in,
                              void* d_out, int out_size, void* d_ws, size_t ws_size,
                              hipStream_t stream) {
  const float* query = (const float*)d_in[0];
  const float* key_  = (const float*)d_in[1];
  const float* value = (const float*)d_in[2];
  const float* Wa_w  = (const float*)d_in[3];
  const float* Wa_b  = (const float*)d_in[4];
  const float* Ua_w  = (const float*)d_in[5];
  const float* Ua_b  = (const float*)d_in[6];
  const float* V_w   = (const float*)d_in[7];
  const float* V_b   = (const float*)d_in[8];
  const float* fcq_w = (const float*)d_in[9];
  const float* fcq_b = (const float*)d_in[10];
  const float* fck_w = (const float*)d_in[11];
  const float* fck_b = (const float*)d_in[12];
  const float* fcv_w = (const float*)d_in[13];
  const float* fcv_b = (const float*)d_in[14];
  const float* fco_w = (const float*)d_in[15];
  const float* fco_b = (const float*)d_in[16];

  float* out  = (float*)d_out;                       // [B,L,HID]
  float* attn = out + (size_t)BB * LL * HIDD;        // [B,H,L,L]

  const size_t SZ = (size_t)BB * NHH * LL * DD;      // 524288 floats
  float* ws = (float*)d_ws;
  float* Q  = ws;
  float* Kc = Q  + SZ;
  float* Vc = Kc + SZ;
  float* Qe = Vc + SZ;
  float* Ke = Qe + SZ;
  float* X  = Ke + SZ;

  // 1) Q/K/V projections: 1024x512x512, head-split output (2048 tiles / 4 waves).
  wmma_gemm_nt<<<512, 128, 0, stream>>>(query, HIDD, fcq_w, fcq_b, Q,
                                        BB * LL, HIDD, HIDD, 0);
  wmma_gemm_nt<<<512, 128, 0, stream>>>(key_,  HIDD, fck_w, fck_b, Kc,
                                        BB * LL, HIDD, HIDD, 0);
  wmma_gemm_nt<<<512, 128, 0, stream>>>(value, HIDD, fcv_w, fcv_b, Vc,
                                        BB * LL, HIDD, HIDD, 0);

  // 2) Per-head additive projections: 8192x64x64.
  wmma_gemm_nt<<<512, 128, 0, stream>>>(Q,  DD, Wa_w, Wa_b, Qe,
                                        BB * NHH * LL, DD, DD, 1);
  wmma_gemm_nt<<<512, 128, 0, stream>>>(Kc, DD, Ua_w, Ua_b, Ke,
                                        BB * NHH * LL, DD, DD, 1);

  // 3) tanh energy + fused softmax -> attention output region.
  energy_softmax<<<256, 256, 0, stream>>>(Qe, Ke, V_w, V_b, attn);

  // 4) attn @ V -> X in [B,L,HID] layout.
  attn_v<<<512, 128, 0, stream>>>(attn, Vc, X);

  // 5) Output projection: 1024x512x512 -> out region.
  wmma_gemm_nt<<<512, 128, 0, stream>>>(X, HIDD, fco_w, fco_b, out,
                                        BB * LL, HIDD, HIDD, 1);

  (void)in_sizes; (void)n_in; (void)out_size; (void)ws_size;
}